// SegGiniGraphHead_76991583748727
// MI455X (gfx1250) — compile-verified
//
#include <hip/hip_runtime.h>

// ---------------- problem constants (match reference) ----------------
#define NN   100000     // nodes
#define DD   64         // feature dim
#define NG   16         // graphs
#define NC   6          // classes
#define EPS  1e-5f
#define WSTR 65         // padded LDS row stride (bank-conflict avoidance, 64 banks)

typedef float v2f __attribute__((ext_vector_type(2)));
typedef float v8f __attribute__((ext_vector_type(8)));

// ---------------- zero fill ----------------
__global__ __launch_bounds__(256) void k_zero(float* __restrict__ p, int n) {
    int i = blockIdx.x * 256 + threadIdx.x;
    if (i < n) p[i] = 0.0f;
}
__global__ __launch_bounds__(256) void k_zero_i(int* __restrict__ p, int n) {
    int i = blockIdx.x * 256 + threadIdx.x;
    if (i < n) p[i] = 0;
}

// ---------------- CSR build: degree histogram by dst ----------------
__global__ __launch_bounds__(256) void k_deg(const int* __restrict__ dst,
                                             int* __restrict__ deg, int nE) {
    int e = blockIdx.x * 256 + threadIdx.x;
    if (e < nE) atomicAdd(&deg[dst[e]], 1);
}

// ---------------- exclusive prefix sum (single block, chunked Hillis-Steele) ----------------
__global__ __launch_bounds__(1024) void k_scan(const int* __restrict__ deg,
                                               int* __restrict__ rowptr, int n) {
    __shared__ int buf[1024];
    __shared__ int carry;
    const int t = threadIdx.x;
    if (t == 0) carry = 0;
    __syncthreads();
    for (int base = 0; base < n; base += 1024) {
        int i = base + t;
        int v = (i < n) ? deg[i] : 0;
        buf[t] = v;
        __syncthreads();
        for (int off = 1; off < 1024; off <<= 1) {
            int x = (t >= off) ? buf[t - off] : 0;
            __syncthreads();
            buf[t] += x;
            __syncthreads();
        }
        int total = buf[1023];
        int excl = buf[t] - v;
        if (i < n) rowptr[i] = carry + excl;
        __syncthreads();
        if (t == 0) carry += total;
        __syncthreads();
    }
}

// ---------------- CSR fill: csr[rowptr[dst]+k] = src ----------------
__global__ __launch_bounds__(256) void k_fill(const int* __restrict__ src,
                                              const int* __restrict__ dst,
                                              const int* __restrict__ rowptr,
                                              int* __restrict__ cursor,
                                              int* __restrict__ csr, int nE) {
    int e = blockIdx.x * 256 + threadIdx.x;
    if (e >= nE) return;
    int d = dst[e];
    int pos = atomicAdd(&cursor[d], 1);
    csr[rowptr[d] + pos] = src[e];
}

// ---------------- gather aggregation: agg[i] = sum_{e: dst==i} h[src_e] ----------------
// One wave per node; lane owns 2 features; neighbor rows stream as coalesced
// 256B/instruction global_load_b64 (h is L2-resident: 25.6MB << 192MB L2).
__global__ __launch_bounds__(256) void k_agg(const float* __restrict__ h,
                                             const int* __restrict__ rowptr,
                                             const int* __restrict__ deg,
                                             const int* __restrict__ csr,
                                             float* __restrict__ agg, int nNodes) {
    int wid = threadIdx.x >> 5;
    int lane = threadIdx.x & 31;
    int node = blockIdx.x * 8 + wid;
    if (node >= nNodes) return;
    int start = rowptr[node];
    int end = start + deg[node];
    float ax = 0.f, ay = 0.f;
    for (int e = start; e < end; ++e) {
        int s = csr[e];
        const float2 v = *reinterpret_cast<const float2*>(h + (size_t)s * DD + lane * 2);
        ax += v.x;
        ay += v.y;
    }
    float2 o; o.x = ax; o.y = ay;
    *reinterpret_cast<float2*>(agg + (size_t)node * DD + lane * 2) = o;
}

// ---------------- fused GIN MLP layer (one wave = 16 nodes) ----------------
// hdst = relu(BN2( relu(BN1( (hsrc+agg) @ W1 + b1 )) @ W2 + b2 ))
// fp32 WMMA: V_WMMA_F32_16X16X4_F32, K accumulated over 16 steps of 4.
__global__ __launch_bounds__(32) void k_gin_mlp(
    const float* __restrict__ hsrc, const float* __restrict__ agg,
    float* __restrict__ hdst,
    const float* __restrict__ W1, const float* __restrict__ b1,
    const float* __restrict__ g1, const float* __restrict__ be1,
    const float* __restrict__ m1, const float* __restrict__ v1,
    const float* __restrict__ W2, const float* __restrict__ b2,
    const float* __restrict__ g2, const float* __restrict__ be2,
    const float* __restrict__ m2, const float* __restrict__ v2,
    int nNodes) {
    __shared__ float w1s[DD * WSTR];
    __shared__ float w2s[DD * WSTR];
    __shared__ float tile[16 * WSTR];

    const int lane  = threadIdx.x;      // 0..31 (wave32)
    const int node0 = blockIdx.x * 16;
    if (node0 >= nNodes) return;        // uniform: EXEC stays all-ones for WMMA
    const int hh  = lane >> 4;          // half-wave: 0/1
    const int l16 = lane & 15;

    // stage W1/W2 into padded LDS (coalesced float4 global loads)
    for (int i = lane; i < (DD * DD) / 4; i += 32) {
        int r = i >> 4;
        int c = (i & 15) << 2;
        float4 a = *reinterpret_cast<const float4*>(W1 + r * DD + c);
        float4 b = *reinterpret_cast<const float4*>(W2 + r * DD + c);
        float* p1 = &w1s[r * WSTR + c];
        p1[0] = a.x; p1[1] = a.y; p1[2] = a.z; p1[3] = a.w;
        float* p2 = &w2s[r * WSTR + c];
        p2[0] = b.x; p2[1] = b.y; p2[2] = b.z; p2[3] = b.w;
    }
    // stage z = hsrc + agg for this 16-node tile
    for (int i = lane; i < (16 * DD) / 4; i += 32) {
        int r = i >> 4;
        int c = (i & 15) << 2;
        const size_t off = (size_t)(node0 + r) * DD + c;
        float4 a = *reinterpret_cast<const float4*>(hsrc + off);
        float4 g = *reinterpret_cast<const float4*>(agg + off);
        float* p = &tile[r * WSTR + c];
        p[0] = a.x + g.x; p[1] = a.y + g.y; p[2] = a.z + g.z; p[3] = a.w + g.w;
    }
    __syncthreads();

    // A fragments (16x4 f32 per step): lanes 0-15 hold K0/K1, lanes 16-31 K2/K3
    v2f afrag[16];
#pragma unroll
    for (int kb = 0; kb < 16; ++kb) {
        int k0 = 4 * kb + 2 * hh;
        afrag[kb].x = tile[l16 * WSTR + k0];
        afrag[kb].y = tile[l16 * WSTR + k0 + 1];
    }

    // ---- GEMM1 + bias + BN1 + ReLU -> back into tile ----
#pragma unroll
    for (int n = 0; n < 4; ++n) {
        v8f c = {0.f, 0.f, 0.f, 0.f, 0.f, 0.f, 0.f, 0.f};
#pragma unroll
        for (int kb = 0; kb < 16; ++kb) {
            int k0 = 4 * kb + 2 * hh;
            v2f b;
            b.x = w1s[k0 * WSTR + n * 16 + l16];
            b.y = w1s[(k0 + 1) * WSTR + n * 16 + l16];
            c = __builtin_amdgcn_wmma_f32_16x16x4_f32(false, afrag[kb], false, b,
                                                      (short)0, c, false, false);
        }
        int col = n * 16 + l16;
        float scale = g1[col] / sqrtf(v1[col] + EPS);
        float shift = (b1[col] - m1[col]) * scale + be1[col];
#pragma unroll
        for (int j = 0; j < 8; ++j) {
            float t = c[j] * scale + shift;
            t = t > 0.f ? t : 0.f;
            tile[(j + 8 * hh) * WSTR + col] = t;
        }
    }
    __syncthreads();

    // reload A fragments from intermediate activations
#pragma unroll
    for (int kb = 0; kb < 16; ++kb) {
        int k0 = 4 * kb + 2 * hh;
        afrag[kb].x = tile[l16 * WSTR + k0];
        afrag[kb].y = tile[l16 * WSTR + k0 + 1];
    }

    // ---- GEMM2 + bias + BN2 + ReLU -> hdst ----
#pragma unroll
    for (int n = 0; n < 4; ++n) {
        v8f c = {0.f, 0.f, 0.f, 0.f, 0.f, 0.f, 0.f, 0.f};
#pragma unroll
        for (int kb = 0; kb < 16; ++kb) {
            int k0 = 4 * kb + 2 * hh;
            v2f b;
            b.x = w2s[k0 * WSTR + n * 16 + l16];
            b.y = w2s[(k0 + 1) * WSTR + n * 16 + l16];
            c = __builtin_amdgcn_wmma_f32_16x16x4_f32(false, afrag[kb], false, b,
                                                      (short)0, c, false, false);
        }
        int col = n * 16 + l16;
        float scale = g2[col] / sqrtf(v2[col] + EPS);
        float shift = (b2[col] - m2[col]) * scale + be2[col];
#pragma unroll
        for (int j = 0; j < 8; ++j) {
            float t = c[j] * scale + shift;
            t = t > 0.f ? t : 0.f;
            hdst[(size_t)(node0 + j + 8 * hh) * DD + col] = t;
        }
    }
}

// ---------------- hierarchical segment-sum pool over graphs ----------------
__global__ __launch_bounds__(256) void k_pool(const float* __restrict__ h,
                                              const int* __restrict__ batch,
                                              float* __restrict__ sums,
                                              float* __restrict__ cnts, int nNodes) {
    __shared__ float ls[NG * DD];
    __shared__ float lc[NG];
    int t = threadIdx.x;
    for (int i = t; i < NG * DD; i += 256) ls[i] = 0.f;
    if (t < NG) lc[t] = 0.f;
    __syncthreads();

    int node = blockIdx.x * 64 + (t >> 2);
    int fq = (t & 3) << 4;
    if (node < nNodes) {
        int g = batch[node];
#pragma unroll
        for (int i = 0; i < 16; ++i)
            atomicAdd(&ls[g * DD + fq + i], h[(size_t)node * DD + fq + i]);
        if ((t & 3) == 0) atomicAdd(&lc[g], 1.0f);
    }
    __syncthreads();

    for (int i = t; i < NG * DD; i += 256) {
        float v = ls[i];
        if (v != 0.f) unsafeAtomicAdd(&sums[i], v);
    }
    if (t < NG && lc[t] != 0.f) unsafeAtomicAdd(&cnts[t], lc[t]);
}

// ---------------- heads: pooled @ wp + bp, pooled @ ws + bs ----------------
__global__ __launch_bounds__(96) void k_head(const float* __restrict__ sums,
                                             const float* __restrict__ cnts,
                                             const float* __restrict__ wp,
                                             const float* __restrict__ bp,
                                             const float* __restrict__ wsec,
                                             const float* __restrict__ bsec,
                                             float* __restrict__ out) {
    int t = threadIdx.x;
    if (t >= NG * NC) return;
    int g = t / NC, c = t % NC;
    float inv = 1.0f / fmaxf(cnts[g], 1.0f);
    float accP = 0.f, accS = 0.f;
    for (int f = 0; f < DD; ++f) {
        float p = sums[g * DD + f] * inv;
        accP += p * wp[f * NC + c];
        accS += p * wsec[f * NC + c];
    }
    out[g * NC + c] = accP + bp[c];
    out[NG * NC + g * NC + c] = accS + bsec[c];
}

// ---------------- launch ----------------
extern "C" void kernel_launch(void* const* d_in, const int* in_sizes, int n_in,
                              void* d_out, int out_size, void* d_ws, size_t ws_size,
                              hipStream_t stream) {
    const float* x    = (const float*)d_in[0];
    const int*   ei   = (const int*)d_in[1];
    const int*   bat  = (const int*)d_in[2];
    const float* W1   = (const float*)d_in[3];
    const float* b1   = (const float*)d_in[4];
    const float* g1   = (const float*)d_in[5];
    const float* be1  = (const float*)d_in[6];
    const float* m1   = (const float*)d_in[7];
    const float* v1   = (const float*)d_in[8];
    const float* W2   = (const float*)d_in[9];
    const float* b2   = (const float*)d_in[10];
    const float* g2   = (const float*)d_in[11];
    const float* be2  = (const float*)d_in[12];
    const float* m2   = (const float*)d_in[13];
    const float* v2   = (const float*)d_in[14];
    const float* wp   = (const float*)d_in[15];
    const float* bp   = (const float*)d_in[16];
    const float* wsec = (const float*)d_in[17];
    const float* bsec = (const float*)d_in[18];

    const int nNodes = in_sizes[0] / DD;
    const int nE     = in_sizes[1] / 2;
    const int* src = ei;
    const int* dst = ei + nE;

    // ---- workspace layout ----
    float* wsf  = (float*)d_ws;
    float* agg  = wsf;                            // nNodes*DD
    float* hbuf = agg + (size_t)nNodes * DD;      // nNodes*DD
    float* sums = hbuf + (size_t)nNodes * DD;     // NG*DD
    float* cnts = sums + NG * DD;                 // NG
    int* ibase  = (int*)(cnts + NG);
    int* deg    = ibase;                          // nNodes
    int* rowp   = deg + nNodes;                   // nNodes
    int* cursor = rowp + nNodes;                  // nNodes
    int* csr    = cursor + nNodes;                // nE

    const int nodeBlks = (nNodes + 255) / 256;
    const int edgeBlks = (nE + 255) / 256;
    const int tileBlks = (nNodes + 15) / 16;
    const int aggBlks  = (nNodes + 7) / 8;
    const int poolBlks = (nNodes + 63) / 64;

    // ---- CSR build (once per call; amortized over the 3 layers) ----
    k_zero_i<<<nodeBlks, 256, 0, stream>>>(deg, nNodes);
    k_zero_i<<<nodeBlks, 256, 0, stream>>>(cursor, nNodes);
    k_deg<<<edgeBlks, 256, 0, stream>>>(dst, deg, nE);
    k_scan<<<1, 1024, 0, stream>>>(deg, rowp, nNodes);
    k_fill<<<edgeBlks, 256, 0, stream>>>(src, dst, rowp, cursor, csr, nE);

    // ---- 3 GIN layers ----
    for (int layer = 0; layer < 3; ++layer) {
        const float* hsrc = (layer == 0) ? x : hbuf;
        k_agg<<<aggBlks, 256, 0, stream>>>(hsrc, rowp, deg, csr, agg, nNodes);
        const int wo = layer * DD * DD, po = layer * DD;
        k_gin_mlp<<<tileBlks, 32, 0, stream>>>(
            hsrc, agg, hbuf,
            W1 + wo, b1 + po, g1 + po, be1 + po, m1 + po, v1 + po,
            W2 + wo, b2 + po, g2 + po, be2 + po, m2 + po, v2 + po,
            nNodes);
    }

    // ---- pooling + heads ----
    k_zero<<<(NG * DD + NG + 255) / 256, 256, 0, stream>>>(sums, NG * DD + NG);
    k_pool<<<poolBlks, 256, 0, stream>>>(hbuf, bat, sums, cnts, nNodes);
    k_head<<<1, 96, 0, stream>>>(sums, cnts, wp, bp, wsec, bsec, (float*)d_out);
}